// OneHotProjEmbedding_2869038153927
// MI455X (gfx1250) — compile-verified
//
#include <hip/hip_runtime.h>

// OneHotProjEmbedding on gfx1250 (MI455X).
//
// out[t, :] = W[:, idx[t]] + b   -- a pure gather, memory bound.
//   writes: 32 MB, reads: ~0.25 GB (random 64B lines, L2-cached) => ~15 us @ 23.3 TB/s.
//
// CDNA5 usage:
//  * v_wmma_f32_16x16x4_f32 performs the bias add bit-exactly:
//      A[t][0]=1 (else 0), B[0][e]=bias[e]  =>  D = 1*bias + 0 + 0 + 0 + C = C + bias.
//    The gather is written straight into the 16x16 f32 C-tile register layout
//    (lane = column N, VGPR r = row M, lanes 16..31 hold M+8).
//  * global_load_async_to_lds_b32 (+ s_wait_asynccnt) stages the block's 128
//    token indices and the 128-entry bias vector into LDS.

typedef float v8f __attribute__((ext_vector_type(8)));
typedef float v2f __attribute__((ext_vector_type(2)));

#define THREADS 256
#define TOK_PER_BLOCK 128   // 8 waves * 16 tokens
#define EMBED 128

__global__ __launch_bounds__(THREADS)
void OneHotProjEmbedding_kernel(const int* __restrict__ idx,     // [n_tokens] int32
                                const float* __restrict__ W,     // [EMBED, num_labels]
                                const float* __restrict__ bias,  // [EMBED]
                                float* __restrict__ out,         // [n_tokens, EMBED]
                                int num_labels,
                                long long n_tokens)
{
    __shared__ int   s_idx[TOK_PER_BLOCK];
    __shared__ float s_bias[EMBED];

    const int tid = threadIdx.x;
    const long long base = (long long)blockIdx.x * TOK_PER_BLOCK;

    // ---- Stage indices + bias into LDS with gfx1250 async global->LDS loads ----
    {
        unsigned lds_addr;
        unsigned long long gaddr;
        if (tid < TOK_PER_BLOCK) {
            long long t = base + tid;
            if (t >= n_tokens) t = n_tokens - 1;     // clamp for tail blocks
            gaddr    = (unsigned long long)(const void*)(idx + t);
            lds_addr = (unsigned)(unsigned long long)&s_idx[tid];
        } else {
            int e = tid - TOK_PER_BLOCK;             // 0..127
            gaddr    = (unsigned long long)(const void*)(bias + e);
            lds_addr = (unsigned)(unsigned long long)&s_bias[e];
        }
        asm volatile("global_load_async_to_lds_b32 %0, %1, off"
                     :: "v"(lds_addr), "v"(gaddr) : "memory");
        asm volatile("s_wait_asynccnt 0" ::: "memory");
    }
    __syncthreads();

    if (base + TOK_PER_BLOCK <= n_tokens) {
        // ---------------- full block: WMMA-tile path (EXEC all ones) ----------------
        const int lane = tid & 31;
        const int wave = tid >> 5;     // 0..7
        const int col  = lane & 15;    // tile column N
        const int half = lane >> 4;    // 0: rows 0..7, 1: rows 8..15

        const long long tok0 = base + (long long)wave * 16;   // this wave's 16 tokens

        // per-lane token indices for its 8 C/D rows
        int myIdx[8];
#pragma unroll
        for (int r = 0; r < 8; ++r)
            myIdx[r] = s_idx[wave * 16 + r + 8 * half];

        // A (16x4 f32): A[M][0] = 1, else 0.
        // Layout: VGPR0 = K0 (lanes 0-15) / K2 (lanes 16-31); VGPR1 = K1/K3.
        v2f A;
        A.x = (half == 0) ? 1.0f : 0.0f;
        A.y = 0.0f;

#pragma unroll 2
        for (int n = 0; n < 8; ++n) {               // 8 tiles of 16 embed columns
            const int e = n * 16 + col;             // this lane's embed row of W
            const float* rowBase = W + (long long)e * num_labels;

            union { v8f v; float f[8]; } c, d;
#pragma unroll
            for (int r = 0; r < 8; ++r)             // 8 scalar gathers per lane
                c.f[r] = rowBase[myIdx[r]];

            // B (4x16 f32): B[0][N] = bias[e]; K1..K3 don't-care (A is 0 there).
            v2f Bop;
            Bop.x = s_bias[e];
            Bop.y = 0.0f;

#if __has_builtin(__builtin_amdgcn_wmma_f32_16x16x4_f32)
            d.v = __builtin_amdgcn_wmma_f32_16x16x4_f32(
                      false, A, false, Bop, (short)0, c.v, false, false);
#else
#pragma unroll
            for (int r = 0; r < 8; ++r) d.f[r] = c.f[r] + Bop.x;
#endif

            // store: per r, lanes 0-15 write 64B contiguous, lanes 16-31 likewise
            float* o = out + (tok0 + 8 * half) * EMBED + e;
#pragma unroll
            for (int r = 0; r < 8; ++r)
                o[(long long)r * EMBED] = d.f[r];
        }
    } else {
        // ---------------- tail (never taken at B*S=65536): scalar path ----------------
        for (long long t = base; t < n_tokens; ++t) {
            int id = idx[t];
            for (int e = tid; e < EMBED; e += THREADS)
                out[t * EMBED + e] = W[(long long)e * num_labels + id] + bias[e];
        }
    }
}

extern "C" void kernel_launch(void* const* d_in, const int* in_sizes, int n_in,
                              void* d_out, int out_size, void* d_ws, size_t ws_size,
                              hipStream_t stream) {
    (void)n_in; (void)d_ws; (void)ws_size; (void)out_size;
    const int*   X  = (const int*)d_in[0];    // [B,S,1] int32
    const float* W  = (const float*)d_in[1];  // [EMBED, num_labels]
    const float* b  = (const float*)d_in[2];  // [EMBED]
    float* out = (float*)d_out;

    const long long n_tokens  = in_sizes[0];           // B*S
    const int embed           = in_sizes[2];           // 128
    const int num_labels      = in_sizes[1] / embed;   // 500000

    const int blocks = (int)((n_tokens + TOK_PER_BLOCK - 1) / TOK_PER_BLOCK);
    OneHotProjEmbedding_kernel<<<blocks, THREADS, 0, stream>>>(
        X, W, b, out, num_labels, n_tokens);
}